// AttentionAggregator_42477226557515
// MI455X (gfx1250) — compile-verified
//
#include <hip/hip_runtime.h>
#include <hip/hip_bf16.h>
#include <math.h>

typedef float v2f __attribute__((ext_vector_type(2)));
typedef float v8f __attribute__((ext_vector_type(8)));

#define FDIM   256
#define NSAMP  10
#define TILE   16
#define NTHR   160   // 5 waves of 32 -> slots 0..9, EXEC all-ones per wave

// One workgroup = 16 nodes. 5 full waves; wave w owns sample slots 2w, 2w+1.
__launch_bounds__(NTHR)
__global__ void attn_agg_kernel(const float* __restrict__ features,
                                const int*  __restrict__ nodes,
                                const int*  __restrict__ unique_ids,
                                const int*  __restrict__ neigh_idx,
                                float* __restrict__ out) {
  __shared__ float Qs[TILE * FDIM];        // 16 KB query tile
  __shared__ int   vidS[TILE * NSAMP];     // vocab row id per (node,slot)
  __shared__ int   colS[TILE * NSAMP];     // column index (for dedupe)
  __shared__ float scS[TILE * NSAMP];      // raw scores
  __shared__ float wS [TILE * NSAMP];      // softmax weights (0 for dup slots)

  const int base = blockIdx.x * TILE;
  const int tid  = threadIdx.x;

  // ---- Stage A: stage queries in LDS; gather neighbor ids ----------------
  for (int v = tid; v < TILE * FDIM; v += NTHR) {
    const int row = v >> 8;               // FDIM == 256
    const int f   = v & (FDIM - 1);
    const int nid = nodes[base + row];
    Qs[v] = features[(size_t)nid * FDIM + f];
  }
  if (tid < TILE * NSAMP) {
    const int i = tid / NSAMP, s = tid - i * NSAMP;
    const int col = neigh_idx[(base + i) * NSAMP + s];
    colS[tid] = col;
    vidS[tid] = unique_ids[col];
  }
  __syncthreads();

  // ---- Stage B: scores via V_WMMA_F32_16X16X4_F32 ------------------------
  const int lane = tid & 31;
  const int wav  = tid >> 5;
  const int s0   = 2 * wav, s1 = 2 * wav + 1;
  const int n    = lane & 15;                    // matrix row/col this lane feeds
  const int koff = (lane < 16) ? 0 : 2;          // K sub-offset per ISA layout

  const float* __restrict__ b0row = features + (size_t)vidS[n * NSAMP + s0] * FDIM;
  const float* __restrict__ b1row = features + (size_t)vidS[n * NSAMP + s1] * FDIM;
  const float* arow = &Qs[n * FDIM];

  v8f acc0 = {0.f, 0.f, 0.f, 0.f, 0.f, 0.f, 0.f, 0.f};
  v8f acc1 = {0.f, 0.f, 0.f, 0.f, 0.f, 0.f, 0.f, 0.f};
#pragma unroll 8
  for (int k0 = 0; k0 < FDIM; k0 += 4) {
    v2f a  = *(const v2f*)(arow  + k0 + koff);   // ds_load_b64
    v2f b0 = *(const v2f*)(b0row + k0 + koff);   // global_load_b64 (L2-resident)
    v2f b1 = *(const v2f*)(b1row + k0 + koff);
    acc0 = __builtin_amdgcn_wmma_f32_16x16x4_f32(false, a, false, b0,
                                                 (short)0, acc0, false, false);
    acc1 = __builtin_amdgcn_wmma_f32_16x16x4_f32(false, a, false, b1,
                                                 (short)0, acc1, false, false);
  }

  // Diagonal of the 16x16 C tile: VGPR r @ lane r (M=0..7), VGPR r @ lane r+24 (M=8..15)
  const bool hasDiag = (lane < 8) || (lane >= 24);
  const int  r       = (lane < 8) ? lane : (lane - 24);
  const int  node    = (lane < 8) ? lane : (lane - 16);
  float d0 = acc0[0], d1 = acc1[0];
#pragma unroll
  for (int rr = 1; rr < 8; ++rr) {
    d0 = (r == rr) ? acc0[rr] : d0;
    d1 = (r == rr) ? acc1[rr] : d1;
  }
  if (hasDiag) {
    scS[node * NSAMP + s0] = d0;
    scS[node * NSAMP + s1] = d1;
  }
  __syncthreads();

  // ---- Stage C: dedupe (on column index) + softmax, 1 thread per node ----
  if (tid < TILE) {
    const int i = tid;
    float sc[NSAMP], w[NSAMP];
    int   cl[NSAMP];
    bool  vld[NSAMP];
    float m = -INFINITY;
#pragma unroll
    for (int s = 0; s < NSAMP; ++s) {
      cl[s] = colS[i * NSAMP + s];
      sc[s] = scS[i * NSAMP + s];
      bool dup = false;
#pragma unroll
      for (int t = 0; t < NSAMP; ++t)
        if (t < s && cl[t] == cl[s]) dup = true;
      vld[s] = !dup;
      if (vld[s]) m = fmaxf(m, sc[s]);
    }
    float sum = 0.f;
#pragma unroll
    for (int s = 0; s < NSAMP; ++s) {
      w[s] = vld[s] ? expf(sc[s] - m) : 0.f;
      sum += w[s];
    }
    const float inv = 1.f / sum;
#pragma unroll
    for (int s = 0; s < NSAMP; ++s)
      wS[i * NSAMP + s] = w[s] * inv;
  }
  __syncthreads();

  // ---- Stage D: weighted aggregation (exact VALU, coalesced-ish) ---------
  for (int e = tid; e < TILE * FDIM; e += NTHR) {
    const int i = e >> 8;
    const int f = e & (FDIM - 1);
    float acc = 0.f;
#pragma unroll
    for (int s = 0; s < NSAMP; ++s) {
      const float wv = wS[i * NSAMP + s];
      if (wv != 0.f)
        acc += wv * features[(size_t)vidS[i * NSAMP + s] * FDIM + f];
    }
    out[(size_t)(base + i) * FDIM + f] = acc;
  }
}

extern "C" void kernel_launch(void* const* d_in, const int* in_sizes, int n_in,
                              void* d_out, int out_size, void* d_ws, size_t ws_size,
                              hipStream_t stream) {
  const float* features   = (const float*)d_in[0];
  const int*   nodes      = (const int*)  d_in[1];
  const int*   unique_ids = (const int*)  d_in[2];
  const int*   neigh_idx  = (const int*)  d_in[3];
  float*       out        = (float*)      d_out;

  const int N = in_sizes[1];            // 4096 nodes, multiple of TILE
  dim3 grid(N / TILE), block(NTHR);
  attn_agg_kernel<<<grid, block, 0, stream>>>(features, nodes, unique_ids,
                                              neigh_idx, out);
}